// OTKernel_82454782148916
// MI455X (gfx1250) — compile-verified
//
#include <hip/hip_runtime.h>
#include <hip/hip_bf16.h>

typedef float v2f __attribute__((ext_vector_type(2)));
typedef float v8f __attribute__((ext_vector_type(8)));

#define NEG_INF_F (-1e8f)

// ---------------------------------------------------------------------------
// Kernel 0: zero u, v; compute loga[b] = log(O / sum(mask[b,:]))
// ---------------------------------------------------------------------------
__global__ __launch_bounds__(256) void init_kernel(float* __restrict__ u,
                                                   float* __restrict__ v,
                                                   float* __restrict__ loga,
                                                   const unsigned char* __restrict__ mask,
                                                   int S, int O) {
    int b = blockIdx.x;
    int t = threadIdx.x;
    for (int i = t; i < S; i += 256) u[(size_t)b * S + i] = 0.0f;
    for (int i = t; i < O; i += 256) v[(size_t)b * O + i] = 0.0f;
    int cnt = 0;
    for (int i = t; i < S; i += 256) cnt += (mask[(size_t)b * S + i] != 0) ? 1 : 0;
    __shared__ int red[256];
    red[t] = cnt;
    __syncthreads();
    for (int st = 128; st > 0; st >>= 1) {
        if (t < st) red[t] += red[t + st];
        __syncthreads();
    }
    if (t == 0) loga[b] = logf((float)O / (float)red[0]);
}

// ---------------------------------------------------------------------------
// Kernel 1: K[b,s,o] = pf(s,o) * (dot(input[b,s,:], target[b,o,:]) - iou[b,s,o]) / eps
// WMMA f32 16x16x4, D hardcoded 128. Block = 256 threads = 8 waves,
// wave grid 2x4 -> block tile 32(s) x 64(o). A/B staged in LDS.
// ---------------------------------------------------------------------------
__global__ __launch_bounds__(256) void gemm_k_kernel(const float* __restrict__ input,
                                                     const float* __restrict__ target,
                                                     const float* __restrict__ iou,
                                                     float* __restrict__ Kbuf,
                                                     int S, int O,
                                                     float inv_eps, float inv_S, float inv_O,
                                                     float inv_sigma) {
    __shared__ float lA[32 * 128];   // [s_local][k]
    __shared__ float lB[64 * 128];   // [o_local][k]

    const int b  = blockIdx.z;
    const int s0 = blockIdx.y * 32;
    const int o0 = blockIdx.x * 64;
    const int t  = threadIdx.x;

    // Tiles cover full K=128 rows -> contiguous memory regions, float4 copies.
    const float4* Ain = (const float4*)(input  + ((size_t)b * S + s0) * 128);
    const float4* Bin = (const float4*)(target + ((size_t)b * O + o0) * 128);
#pragma unroll
    for (int i = 0; i < 4; ++i)  ((float4*)lA)[t + i * 256] = Ain[t + i * 256];
#pragma unroll
    for (int i = 0; i < 8; ++i)  ((float4*)lB)[t + i * 256] = Bin[t + i * 256];
    __syncthreads();

    const int lane    = t & 31;
    const int w       = t >> 5;
    const int wm      = w >> 2;        // 0..1  -> s sub-tile
    const int wn      = w & 3;         // 0..3  -> o sub-tile
    const int halfsel = lane >> 4;     // 0 or 1 (lane group)
    const int l16     = lane & 15;

    const float* aRow = lA + (wm * 16 + l16) * 128 + 2 * halfsel;
    const float* bRow = lB + (wn * 16 + l16) * 128 + 2 * halfsel;

    v8f acc = {0.f, 0.f, 0.f, 0.f, 0.f, 0.f, 0.f, 0.f};
#pragma unroll
    for (int kk = 0; kk < 128; kk += 4) {
        v2f a  = *(const v2f*)(aRow + kk);   // A[m][2*halfsel + j]
        v2f bb = *(const v2f*)(bRow + kk);   // B[2*halfsel + j][n] (from o-major tile)
        acc = __builtin_amdgcn_wmma_f32_16x16x4_f32(false, a, false, bb,
                                                    (short)0, acc, false, false);
    }

    // C/D layout: VGPR r -> M = r + 8*halfsel, N = l16
    const int oo = o0 + wn * 16 + l16;
    const float oN = (float)oo * inv_O;
#pragma unroll
    for (int r = 0; r < 8; ++r) {
        const int ss = s0 + wm * 16 + r + 8 * halfsel;
        const float tp = ((float)ss * inv_S - oN) * inv_sigma;
        const float pf = __expf(-tp * tp);
        const size_t off = ((size_t)b * S + ss) * O + oo;
        Kbuf[off] = pf * (acc[r] - iou[off]) * inv_eps;
    }
}

// ---------------------------------------------------------------------------
// Kernel 2 (row update): u[b,s] = mask ? loga[b] - u_old - lse_o(K[b,s,:] + v[b,:])
//                                       : NEG_INF
// One wave per row; online logsumexp + wave32 butterfly reduction.
// ---------------------------------------------------------------------------
__global__ __launch_bounds__(256) void row_update_kernel(const float* __restrict__ Kbuf,
                                                         const float* __restrict__ v,
                                                         float* __restrict__ u,
                                                         const float* __restrict__ loga,
                                                         const unsigned char* __restrict__ mask,
                                                         int S, int O) {
    const int rowg = blockIdx.x * 8 + (threadIdx.x >> 5);  // global row = b*S + s
    const int lane = threadIdx.x & 31;
    const int b = rowg / S;
    const float* Krow = Kbuf + (size_t)rowg * O;
    const float* vb   = v + (size_t)b * O;

    float m = -3.4e38f, sum = 0.0f;
    for (int o = lane; o < O; o += 32) {
        const float x = Krow[o] + vb[o];
        const float mn = fmaxf(m, x);
        sum = sum * __expf(m - mn) + __expf(x - mn);
        m = mn;
    }
#pragma unroll
    for (int off = 16; off > 0; off >>= 1) {
        const float m2 = __shfl_xor(m, off);
        const float s2 = __shfl_xor(sum, off);
        const float mn = fmaxf(m, m2);
        sum = sum * __expf(m - mn) + s2 * __expf(m2 - mn);
        m = mn;
    }
    if (lane == 0) {
        const float lse = m + logf(sum);
        float un = loga[b] - u[rowg] - lse;
        if (mask[rowg] == 0) un = NEG_INF_F;
        u[rowg] = un;
    }
}

// ---------------------------------------------------------------------------
// Kernel 3 (col update): v[b,o] = -(v_old + lse_s(K[b,:,o] + u[b,:]))
// Block (32 o-lanes, 8 s-threads): every wave read is 128B contiguous.
// ---------------------------------------------------------------------------
__global__ __launch_bounds__(256) void col_update_kernel(const float* __restrict__ Kbuf,
                                                         const float* __restrict__ u,
                                                         float* __restrict__ v,
                                                         int S, int O) {
    const int b  = blockIdx.y;
    const int o  = blockIdx.x * 32 + threadIdx.x;
    const int ty = threadIdx.y;
    const float* Kb = Kbuf + (size_t)b * S * O;
    const float* ub = u + (size_t)b * S;

    float m = -3.4e38f, sum = 0.0f;
    for (int s = ty; s < S; s += 8) {
        const float x = Kb[(size_t)s * O + o] + ub[s];
        const float mn = fmaxf(m, x);
        sum = sum * __expf(m - mn) + __expf(x - mn);
        m = mn;
    }
    __shared__ float sm[8][32];
    __shared__ float ss[8][32];
    sm[ty][threadIdx.x] = m;
    ss[ty][threadIdx.x] = sum;
    __syncthreads();
    if (ty == 0) {
#pragma unroll
        for (int j = 1; j < 8; ++j) {
            const float m2 = sm[j][threadIdx.x];
            const float s2 = ss[j][threadIdx.x];
            const float mn = fmaxf(m, m2);
            sum = sum * __expf(m - mn) + s2 * __expf(m2 - mn);
            m = mn;
        }
        const float lse = m + logf(sum);
        const size_t vi = (size_t)b * O + o;
        v[vi] = -(v[vi] + lse);
    }
}

// ---------------------------------------------------------------------------
// Kernel 4: out[b,o,s] = pf(s,o) * exp(K[b,s,o] + u[b,s] + v[b,o]) * S
// 32x32 LDS transpose: coalesced K reads (along o), coalesced out writes (along s).
// ---------------------------------------------------------------------------
__global__ __launch_bounds__(256) void output_kernel(const float* __restrict__ Kbuf,
                                                     const float* __restrict__ u,
                                                     const float* __restrict__ v,
                                                     float* __restrict__ out,
                                                     int S, int O,
                                                     float inv_S, float inv_O,
                                                     float inv_sigma) {
    __shared__ float tile[32][33];
    const int b  = blockIdx.z;
    const int o0 = blockIdx.y * 32;
    const int s0 = blockIdx.x * 32;
    const int x = threadIdx.x, y = threadIdx.y;

    const float* Kb = Kbuf + (size_t)b * S * O;
    const float* ub = u + (size_t)b * S;
    const float* vb = v + (size_t)b * O;

    const int o = o0 + x;
    const float vo = vb[o];
    const float oN = (float)o * inv_O;
    const float scaleS = (float)S;
#pragma unroll
    for (int j = 0; j < 4; ++j) {
        const int sl = y * 4 + j;
        const int s = s0 + sl;
        const float tp = ((float)s * inv_S - oN) * inv_sigma;
        const float pf = __expf(-tp * tp);
        tile[sl][x] = pf * __expf(Kb[(size_t)s * O + o] + ub[s] + vo) * scaleS;
    }
    __syncthreads();
#pragma unroll
    for (int j = 0; j < 4; ++j) {
        const int ol = y * 4 + j;
        out[((size_t)b * O + o0 + ol) * S + s0 + x] = tile[x][ol];
    }
}

// ---------------------------------------------------------------------------
// Host launcher
// ---------------------------------------------------------------------------
extern "C" void kernel_launch(void* const* d_in, const int* in_sizes, int n_in,
                              void* d_out, int out_size, void* d_ws, size_t ws_size,
                              hipStream_t stream) {
    const float*         input  = (const float*)d_in[0];         // B x S x D
    const float*         target = (const float*)d_in[1];         // B x O x D
    const float*         iou    = (const float*)d_in[2];         // B x S x O
    const unsigned char* mask   = (const unsigned char*)d_in[3]; // B x S (bool)
    float*               out    = (float*)d_out;                 // B x O x S

    const long n0 = in_sizes[0], n1 = in_sizes[1], n2 = in_sizes[2], n3 = in_sizes[3];
    const int D = (int)(n0 / n3);            // 128
    const int O = (int)(n2 / n3);            // 512
    const int B = (int)(n1 / ((long)O * D)); // 16
    const int S = (int)(n3 / B);             // 4096
    (void)D; (void)out_size; (void)ws_size; (void)n_in;

    const float inv_eps   = 1.0f / 0.1f;  // 1/EPS
    const float inv_sigma = 1.0f / 0.1f;  // 1/SIGMA
    const float inv_S = 1.0f / (float)S;
    const float inv_O = 1.0f / (float)O;

    // Workspace layout (floats): K[B*S*O] | u[B*S] | v[B*O] | loga[B]
    float* Kbuf = (float*)d_ws;
    const size_t nK = (size_t)B * S * O;
    float* u    = Kbuf + nK;
    float* v    = u + (size_t)B * S;
    float* loga = v + (size_t)B * O;

    // Init u, v, loga
    init_kernel<<<B, 256, 0, stream>>>(u, v, loga, mask, S, O);

    // K = pf * (input @ target^T - iou) / eps   (WMMA f32 16x16x4, L2-resident K)
    dim3 gGrid(O / 64, S / 32, B);
    gemm_k_kernel<<<gGrid, 256, 0, stream>>>(input, target, iou, Kbuf,
                                             S, O, inv_eps, inv_S, inv_O, inv_sigma);

    // 10 Sinkhorn iterations over L2-resident K
    const dim3 cGrid(O / 32, B), cBlk(32, 8);
    for (int it = 0; it < 10; ++it) {
        row_update_kernel<<<(B * S) / 8, 256, 0, stream>>>(Kbuf, v, u, loga, mask, S, O);
        col_update_kernel<<<cGrid, cBlk, 0, stream>>>(Kbuf, u, v, S, O);
    }

    // out = transpose(pf * exp(K + u + v) * S)
    dim3 oGrid(S / 32, O / 32, B), oBlk(32, 8);
    output_kernel<<<oGrid, oBlk, 0, stream>>>(Kbuf, u, v, out, S, O,
                                              inv_S, inv_O, inv_sigma);
}